// LocalGOCor_40209483825172
// MI455X (gfx1250) — compile-verified
//
#include <hip/hip_runtime.h>
#include <stdint.h>

typedef float v2f __attribute__((ext_vector_type(2)));
typedef float v8f __attribute__((ext_vector_type(8)));

#define B_     16
#define C_     128
#define H_     128
#define W_     128
#define HW_    (H_ * W_)
#define RAD    4
#define ND     9                 // 2*RAD+1 displacements per axis
#define YT     8                 // output rows per block (one wave per row)
#define XT     16                // output cols per block (WMMA M)
#define QROWS  (YT + 2 * RAD)    // 16 query rows resident in LDS
#define QCOLS  24                // exactly the needed cols (x0-4 .. x0+19)
#define QPITCH (QCOLS * 4)       // floats per (kk,row) slab: [col][c%4]
#define GP     34                // diag-scratch pitch (bank-conflict free)

// LDS q layout: [c/4][row][col][c%4]  -> one b64 B-fragment load, no conflicts
__global__ __launch_bounds__(256) void gocor_corr_wmma(
    const float* __restrict__ ref,
    const float* __restrict__ qry,
    const float* __restrict__ gainp,
    float* __restrict__ out)
{
  __shared__ float qlds[32 * QROWS * QPITCH + 128];  // +pad: N-tile1 col overrun
  __shared__ float gbuf[8 * 16 * GP];                // per-wave Gram scratch

  const int tid  = threadIdx.x;
  const int lane = tid & 31;
  const int wave = tid >> 5;
  const int x0 = blockIdx.x * XT;
  const int y0 = blockIdx.y * YT;
  const int b  = blockIdx.z;
  const float gain = gainp[0];

  const size_t qbase = (size_t)b * C_ * HW_;
  const bool interior = (x0 >= RAD) && (x0 + XT + RAD <= W_) &&
                        (y0 >= RAD) && (y0 + YT + RAD <= H_);

  // async copy one query element global -> transposed LDS slot (ASYNCcnt path)
  auto issue_async = [&](int c, int rr, int cl) {
    uint64_t ga = (uint64_t)(const void*)(qry + qbase + (size_t)c * HW_ +
                                          (size_t)(y0 - RAD + rr) * W_ +
                                          (x0 - RAD + cl));
    uint32_t la = (uint32_t)(size_t)(const void*)
        (&qlds[((c >> 2) * QROWS + rr) * QPITCH + cl * 4 + (c & 3)]);
    asm volatile("global_load_async_to_lds_b32 %0, %1, off"
                 :: "v"(la), "v"(ga) : "memory");
  };

  if (interior) {
    // ---- async prefill of rows 0..7 (rows 8..15 pipelined into dy loop) ----
    for (int it = 0; it < (C_ * YT * QCOLS) / 256; ++it) {  // 96 iters
      int flat = it * 256 + tid;
      int c    = flat / (YT * QCOLS);
      int rem  = flat - c * (YT * QCOLS);
      int rr   = rem / QCOLS;
      int cl   = rem - rr * QCOLS;
      issue_async(c, rr, cl);
    }
  } else {
    // ---- boundary: synchronous predicated fill with zero padding ----
    for (int it = 0; it < (C_ * QROWS * QCOLS) / 256; ++it) {  // 192 iters
      int flat = it * 256 + tid;
      int c    = flat / (QROWS * QCOLS);
      int rem  = flat - c * (QROWS * QCOLS);
      int rr   = rem / QCOLS;
      int cl   = rem - rr * QCOLS;
      int gy = y0 - RAD + rr;
      int gx = x0 - RAD + cl;
      float v = 0.0f;
      if (gy >= 0 && gy < H_ && gx >= 0 && gx < W_)
        v = qry[qbase + (size_t)c * HW_ + (size_t)gy * W_ + gx];
      qlds[((c >> 2) * QROWS + rr) * QPITCH + cl * 4 + (c & 3)] = v;
    }
  }

  // ---- A fragments: ref features for this wave's row, full K=128 ----
  // 32-bit A 16x4 layout: lanes 0-15 hold K=k0,k0+1 ; lanes 16-31 hold K=k0+2,k0+3
  const int xi = lane & 15;
  const int cm = (lane < 16) ? 0 : 2;
  const int y  = y0 + wave;
  const size_t rbase = (size_t)b * C_ * HW_ + (size_t)y * W_ + x0 + xi;
  v2f a[32];
#pragma unroll
  for (int kk = 0; kk < 32; ++kk) {
    int c0 = 4 * kk + cm;
    a[kk].x = ref[rbase + (size_t)c0 * HW_];
    a[kk].y = ref[rbase + (size_t)(c0 + 1) * HW_];
  }

  asm volatile("s_wait_asynccnt 0x0" ::: "memory");
  __syncthreads();

  float* gb = &gbuf[wave * 16 * GP];

  for (int dy = 0; dy < ND; ++dy) {
    // ---- pipelined async fill of row 8+dy (first needed at iteration dy+1) ----
    if (interior && dy < YT) {
      int rr = YT + dy;
      for (int it = 0; it < (C_ * QCOLS) / 256; ++it) {  // 12 iters
        int flat = it * 256 + tid;
        int c  = flat / QCOLS;
        int cl = flat - c * QCOLS;
        issue_async(c, rr, cl);
      }
    }

    const int rr = wave + dy;  // query row in LDS strip
    v8f acc0 = {};
    v8f acc1 = {};
#pragma unroll
    for (int kk = 0; kk < 32; ++kk) {
      const int base = (kk * QROWS + rr) * QPITCH;
      const float2* bp0 = (const float2*)&qlds[base + (0 * 16 + xi) * 4 + cm];
      const float2* bp1 = (const float2*)&qlds[base + (1 * 16 + xi) * 4 + cm];
      v2f b0; b0.x = bp0->x; b0.y = bp0->y;
      v2f b1; b1.x = bp1->x; b1.y = bp1->y;
      acc0 = __builtin_amdgcn_wmma_f32_16x16x4_f32(false, a[kk], false, b0,
                                                   (short)0, acc0, false, false);
      acc1 = __builtin_amdgcn_wmma_f32_16x16x4_f32(false, a[kk], false, b1,
                                                   (short)0, acc1, false, false);
    }

    // ---- spill Gram tiles to LDS scratch (C/D layout: M=r(+8), N=lane%16) ----
    const int m0 = (lane < 16) ? 0 : 8;
#pragma unroll
    for (int r = 0; r < 8; ++r) {
      gb[(m0 + r) * GP +  0 + xi] = acc0[r];
      gb[(m0 + r) * GP + 16 + xi] = acc1[r];
    }

    // ---- extract 9 diagonals, scale, streaming (non-temporal) store ----
#pragma unroll
    for (int k = 0; k < 5; ++k) {
      int idx = k * 32 + lane;
      if (idx < ND * 16) {
        int dx = idx >> 4;
        int ii = idx & 15;
        float v = gain * gb[ii * GP + ii + dx];
        size_t o = (((size_t)b * (ND * ND) + (dy * ND + dx)) * H_ + y) * W_ + x0 + ii;
        __builtin_nontemporal_store(v, &out[o]);
      }
    }

    // ---- retire this iteration's async row before anyone consumes it ----
    if (dy + 1 < ND) {
      asm volatile("s_wait_asynccnt 0x0" ::: "memory");
      __syncthreads();
    }
  }
}

extern "C" void kernel_launch(void* const* d_in, const int* in_sizes, int n_in,
                              void* d_out, int out_size, void* d_ws, size_t ws_size,
                              hipStream_t stream) {
  (void)in_sizes; (void)n_in; (void)out_size; (void)d_ws; (void)ws_size;
  const float* ref  = (const float*)d_in[0];
  const float* qry  = (const float*)d_in[1];
  const float* gain = (const float*)d_in[2];
  float* out = (float*)d_out;
  dim3 grid(W_ / XT, H_ / YT, B_);  // (8, 16, 16)
  gocor_corr_wmma<<<grid, 256, 0, stream>>>(ref, qry, gain, out);
}